// Top2Gate_2216203125408
// MI455X (gfx1250) — compile-verified
//
#include <hip/hip_runtime.h>
#include <hip/hip_bf16.h>

typedef float v2f __attribute__((ext_vector_type(2)));
typedef float v4f __attribute__((ext_vector_type(4)));
typedef float v8f __attribute__((ext_vector_type(8)));

#define NEG_INF_F (-1e30f)
#define F32_EPS 1.1920929e-07f

// ---------------------------------------------------------------------------
// Kernel 1: logits = x @ wg.T via V_WMMA_F32_16X16X4_F32
// One wave computes a 16(M) x 64(N=E) strip. Software-pipelined: loads for
// step k+4 are issued before the WMMAs consuming step k, so s_wait_loadcnt
// can retire with loads still outstanding instead of draining to 0.
// ---------------------------------------------------------------------------
__global__ __launch_bounds__(256) void gate_gemm_wmma(
    const float* __restrict__ x, const float* __restrict__ wg,
    float* __restrict__ logits, int T, int D, int E)
{
    const int lane = threadIdx.x & 31;
    const int wave = blockIdx.x * (blockDim.x >> 5) + (threadIdx.x >> 5);
    const int m0 = wave << 4;            // 16 rows per wave
    if (m0 >= T) return;                 // uniform per-wave exit (EXEC stays all-1)

    const int sub   = lane & 15;         // row (A) / col (B) within tile
    const int khalf = (lane >> 4) << 1;  // 0 or 2: which K pair this lane owns

    v8f acc[4] = {};                     // 4 N-tiles of 16 (E = 64)

    const float* __restrict__ xrow = x + (size_t)(m0 + sub) * D + khalf;
    const float* __restrict__ w0 = wg + (size_t)(sub      ) * D + khalf;
    const float* __restrict__ w1 = wg + (size_t)(sub + 16 ) * D + khalf;
    const float* __restrict__ w2 = wg + (size_t)(sub + 32 ) * D + khalf;
    const float* __restrict__ w3 = wg + (size_t)(sub + 48 ) * D + khalf;

    // prologue: stage k = 0
    v2f a  = *(const v2f*)(xrow);
    v2f b0 = *(const v2f*)(w0);
    v2f b1 = *(const v2f*)(w1);
    v2f b2 = *(const v2f*)(w2);
    v2f b3 = *(const v2f*)(w3);

#pragma unroll 4
    for (int k = 4; k < D; k += 4) {
        // issue next-step loads BEFORE consuming current registers
        v2f an  = *(const v2f*)(xrow + k);
        v2f b0n = *(const v2f*)(w0 + k);
        v2f b1n = *(const v2f*)(w1 + k);
        v2f b2n = *(const v2f*)(w2 + k);
        v2f b3n = *(const v2f*)(w3 + k);
        acc[0] = __builtin_amdgcn_wmma_f32_16x16x4_f32(false, a, false, b0, (short)0, acc[0], false, false);
        acc[1] = __builtin_amdgcn_wmma_f32_16x16x4_f32(false, a, false, b1, (short)0, acc[1], false, false);
        acc[2] = __builtin_amdgcn_wmma_f32_16x16x4_f32(false, a, false, b2, (short)0, acc[2], false, false);
        acc[3] = __builtin_amdgcn_wmma_f32_16x16x4_f32(false, a, false, b3, (short)0, acc[3], false, false);
        a = an; b0 = b0n; b1 = b1n; b2 = b2n; b3 = b3n;
    }
    // epilogue: last k-step
    acc[0] = __builtin_amdgcn_wmma_f32_16x16x4_f32(false, a, false, b0, (short)0, acc[0], false, false);
    acc[1] = __builtin_amdgcn_wmma_f32_16x16x4_f32(false, a, false, b1, (short)0, acc[1], false, false);
    acc[2] = __builtin_amdgcn_wmma_f32_16x16x4_f32(false, a, false, b2, (short)0, acc[2], false, false);
    acc[3] = __builtin_amdgcn_wmma_f32_16x16x4_f32(false, a, false, b3, (short)0, acc[3], false, false);

    // C/D layout: VGPR v holds M = v (lanes 0-15) or M = v+8 (lanes 16-31), N = lane&15
    const int moff = (lane >> 4) << 3;   // 0 or 8
#pragma unroll
    for (int v = 0; v < 8; ++v) {
        const size_t M = (size_t)(m0 + v + moff);
#pragma unroll
        for (int n = 0; n < 4; ++n)
            logits[M * E + (n << 4) + sub] = acc[n][v];
    }
}

// ---------------------------------------------------------------------------
// Kernel 2: per-token softmax + top1/top2 (one wave32 per token, 2 experts/lane)
// ---------------------------------------------------------------------------
__global__ __launch_bounds__(256) void softmax_top2(
    const float* __restrict__ logits, float* __restrict__ gates,
    int* __restrict__ idx1, int* __restrict__ idx2,
    float* __restrict__ g1raw, float* __restrict__ g2raw, int T, int E)
{
    const int lane = threadIdx.x & 31;
    const int t = blockIdx.x * (blockDim.x >> 5) + (threadIdx.x >> 5);
    if (t >= T) return;

    const float* __restrict__ row = logits + (size_t)t * E;
    const float l0 = row[lane];
    const float l1 = row[lane + 32];

    float m = fmaxf(l0, l1);
    for (int off = 16; off; off >>= 1) m = fmaxf(m, __shfl_xor(m, off, 32));

    const float e0 = __expf(l0 - m);
    const float e1 = __expf(l1 - m);
    float s = e0 + e1;
    for (int off = 16; off; off >>= 1) s += __shfl_xor(s, off, 32);

    const float g0 = e0 / s;
    const float g1 = e1 / s;
    gates[(size_t)t * E + lane]      = g0;
    gates[(size_t)t * E + lane + 32] = g1;

    // top-1 (prefer lowest index on ties, like jnp.argmax)
    float bv; int bi;
    if (l0 >= l1) { bv = l0; bi = lane; } else { bv = l1; bi = lane + 32; }
    for (int off = 16; off; off >>= 1) {
        float ov = __shfl_xor(bv, off, 32);
        int   oi = __shfl_xor(bi, off, 32);
        if (ov > bv || (ov == bv && oi < bi)) { bv = ov; bi = oi; }
    }
    const int i1 = bi;

    // top-2: argmax with top-1 masked to NEG_INF
    const float ml0 = (lane      == i1) ? NEG_INF_F : l0;
    const float ml1 = (lane + 32 == i1) ? NEG_INF_F : l1;
    if (ml0 >= ml1) { bv = ml0; bi = lane; } else { bv = ml1; bi = lane + 32; }
    for (int off = 16; off; off >>= 1) {
        float ov = __shfl_xor(bv, off, 32);
        int   oi = __shfl_xor(bi, off, 32);
        if (ov > bv || (ov == bv && oi < bi)) { bv = ov; bi = oi; }
    }
    const int i2 = bi;

    float c1 = (lane == i1) ? g0 : ((lane + 32 == i1) ? g1 : 0.f);
    float c2 = (lane == i2) ? g0 : ((lane + 32 == i2) ? g1 : 0.f);
    for (int off = 16; off; off >>= 1) {
        c1 += __shfl_xor(c1, off, 32);
        c2 += __shfl_xor(c2, off, 32);
    }

    if (lane == 0) {
        idx1[t] = i1; idx2[t] = i2;
        g1raw[t] = c1; g2raw[t] = c2;
    }
}

// ---------------------------------------------------------------------------
// Kernel 3: sequential token-order scan per expert (cumsum semantics).
// One block, 64 threads; thread e owns expert e. Arrays are L2-resident.
// ---------------------------------------------------------------------------
__global__ void expert_scan(
    const int* __restrict__ idx1, const int* __restrict__ idx2,
    int* __restrict__ l1s, int* __restrict__ l2s,
    int* __restrict__ cnt1, int T, int C)
{
    const int e = threadIdx.x;
    int c = 0;
    for (int t = 0; t < T; ++t)
        if (idx1[t] == e) { l1s[t] = (c < C) ? c : -1; ++c; }
    cnt1[e] = c;                 // pre-drop count (= T * ce)
    const int off = c;           // loc2 offset = sum(mask1, axis=0)[e]
    int c2 = 0;
    for (int t = 0; t < T; ++t)
        if (idx2[t] == e) { const int p = off + c2; l2s[t] = (p < C) ? p : -1; ++c2; }
}

// ---------------------------------------------------------------------------
// Kernel 4: me[e] = sum_t gates[t,e]  (one block per expert)
// ---------------------------------------------------------------------------
__global__ __launch_bounds__(256) void me_reduce(
    const float* __restrict__ gates, float* __restrict__ me_sum, int T, int E)
{
    __shared__ float sm[256];
    const int e = blockIdx.x;
    float s = 0.f;
    for (int t = threadIdx.x; t < T; t += blockDim.x)
        s += gates[(size_t)t * E + e];
    sm[threadIdx.x] = s;
    __syncthreads();
    for (int w = 128; w > 0; w >>= 1) {
        if ((int)threadIdx.x < w) sm[threadIdx.x] += sm[threadIdx.x + w];
        __syncthreads();
    }
    if (threadIdx.x == 0) me_sum[e] = sm[0];
}

// ---------------------------------------------------------------------------
// Kernel 5: bulk zero of d_out. Touch-once streaming writes -> non-temporal
// (store TH=NT) so ~1.07 GB of zeros doesn't evict L2-resident working data.
// Uses a clang ext_vector (not HIP float4 class) so the builtin accepts it.
// ---------------------------------------------------------------------------
__global__ __launch_bounds__(256) void zero_fill(float* __restrict__ out, size_t n)
{
    const size_t n4 = n >> 2;
    const size_t stride = (size_t)gridDim.x * blockDim.x;
    size_t i = (size_t)blockIdx.x * blockDim.x + threadIdx.x;
    const v4f z = {0.f, 0.f, 0.f, 0.f};
    v4f* __restrict__ out4 = (v4f*)out;
    for (size_t j = i; j < n4; j += stride)
        __builtin_nontemporal_store(z, out4 + j);
    for (size_t j = (n4 << 2) + i; j < n; j += stride)
        __builtin_nontemporal_store(0.f, out + j);
}

// ---------------------------------------------------------------------------
// Kernel 6: scatter the 2 nonzero combine/dispatch entries per token
// ---------------------------------------------------------------------------
__global__ __launch_bounds__(256) void scatter_combine(
    const int* __restrict__ idx1, const int* __restrict__ idx2,
    const int* __restrict__ l1s, const int* __restrict__ l2s,
    const float* __restrict__ g1raw, const float* __restrict__ g2raw,
    float* __restrict__ combine, float* __restrict__ dispatch,
    int T, int E, int C)
{
    const int t = blockIdx.x * blockDim.x + threadIdx.x;
    if (t >= T) return;
    const int p1 = l1s[t], p2 = l2s[t];
    float g1 = (p1 >= 0) ? g1raw[t] : 0.f;
    float g2 = (p2 >= 0) ? g2raw[t] : 0.f;
    const float denom = fmaxf(g1 + g2, F32_EPS);
    g1 /= denom; g2 /= denom;
    const size_t base = (size_t)t * E * C;
    if (p1 >= 0 && g1 != 0.f) {
        const size_t o = base + (size_t)idx1[t] * C + p1;
        combine[o] = g1; dispatch[o] = 1.f;
    }
    if (p2 >= 0 && g2 != 0.f) {
        const size_t o = base + (size_t)idx2[t] * C + p2;
        combine[o] = g2; dispatch[o] = 1.f;
    }
}

// ---------------------------------------------------------------------------
// Kernel 7: l_aux = mean_e(me*ce) * E^2 = sum_e(me*ce) * E
// ---------------------------------------------------------------------------
__global__ void finalize_laux(
    const float* __restrict__ me_sum, const int* __restrict__ cnt1,
    float* __restrict__ out, int T, int E)
{
    __shared__ float sm[64];
    const int e = threadIdx.x;
    const float me = me_sum[e] / (float)T;
    const float ce = (float)cnt1[e] / (float)T;
    sm[e] = me * ce;
    __syncthreads();
    for (int w = 32; w > 0; w >>= 1) {
        if (e < w) sm[e] += sm[e + w];
        __syncthreads();
    }
    if (e == 0) out[0] = sm[0] * (float)E;
}

// ---------------------------------------------------------------------------
extern "C" void kernel_launch(void* const* d_in, const int* in_sizes, int n_in,
                              void* d_out, int out_size, void* d_ws, size_t ws_size,
                              hipStream_t stream)
{
    const float* x  = (const float*)d_in[0];
    const float* wg = (const float*)d_in[1];

    const int E = 64;
    const int D = in_sizes[1] / E;          // 2048
    const int T = in_sizes[0] / D;          // 8192
    const int C = 2 * ((T + E - 1) / E);    // 256

    // workspace layout
    float* logits = (float*)d_ws;                    // T*E
    float* gates  = logits + (size_t)T * E;          // T*E
    float* g1raw  = gates  + (size_t)T * E;          // T
    float* g2raw  = g1raw + T;                       // T
    float* me_sum = g2raw + T;                       // E
    int*   idx1   = (int*)(me_sum + E);              // T
    int*   idx2   = idx1 + T;                        // T
    int*   l1s    = idx2 + T;                        // T
    int*   l2s    = l1s + T;                         // T
    int*   cnt1   = l2s + T;                         // E

    float* out      = (float*)d_out;
    float* combine  = out + 1;
    float* dispatch = combine + (size_t)T * E * C;
    const size_t total_out = (size_t)out_size;

    // 1. bulk zero of output (biggest memory mover)
    zero_fill<<<8192, 256, 0, stream>>>(out, total_out);

    // 2. WMMA gate GEMM: T/16 waves, 8 waves per 256-thread block
    const int gemm_blocks = (T / 16 + 7) / 8;
    gate_gemm_wmma<<<gemm_blocks, 256, 0, stream>>>(x, wg, logits, T, D, E);

    // 3. softmax + top2: 8 tokens (waves) per block
    softmax_top2<<<(T + 7) / 8, 256, 0, stream>>>(logits, gates, idx1, idx2,
                                                  g1raw, g2raw, T, E);

    // 4. sequential per-expert scan (single block, 64 threads)
    expert_scan<<<1, 64, 0, stream>>>(idx1, idx2, l1s, l2s, cnt1, T, C);

    // 5. me reduction (one block per expert)
    me_reduce<<<E, 256, 0, stream>>>(gates, me_sum, T, E);

    // 6. scatter nonzero combine/dispatch entries
    scatter_combine<<<(T + 255) / 256, 256, 0, stream>>>(idx1, idx2, l1s, l2s,
                                                         g1raw, g2raw,
                                                         combine, dispatch, T, E, C);

    // 7. l_aux scalar
    finalize_laux<<<1, 64, 0, stream>>>(me_sum, cnt1, out, T, E);
}